// Model_60705067761855
// MI455X (gfx1250) — compile-verified
//
#include <hip/hip_runtime.h>
#include <hip/hip_bf16.h>

typedef __attribute__((ext_vector_type(2))) float v2f;
typedef __attribute__((ext_vector_type(4))) float f4;
typedef __attribute__((ext_vector_type(8))) float v8f;

#define SEQ   2048
#define BATCH 16
#define HID   1024
#define MTOT  (SEQ * BATCH)   /* 32768 */
#define NTOT  (3 * HID)       /* 3072  */
#define KTOT  HID             /* 1024  */

#define TM 128
#define TN 128
#define TK 32
#define A_STRIDE 36    /* 32 + 4 pad: float4-aligned, conflict-free b64 frag reads */
#define B_STRIDE 144   /* 128 + 16 pad: 144 % 64 == 16 -> disjoint half-wave banks */

// ---------------------------------------------------------------------------
// fp32 GEMM on V_WMMA_F32_16X16X4_F32.  C = A(MxK) * B(KxN), row-major.
// Block: 128x128, 8 waves; wave tile 64x32 = 4x2 subtiles of 16x16.
// ---------------------------------------------------------------------------
__global__ __launch_bounds__(256) void wmma_gemm_f32(
    const float* __restrict__ A, const float* __restrict__ B,
    float* __restrict__ C)
{
  __shared__ float As[TM * A_STRIDE];  // 18 KB
  __shared__ float Bs[TK * B_STRIDE];  // 18 KB

  const int tid  = threadIdx.x;
  const int lane = tid & 31;
  const int wave = tid >> 5;
  const int lm   = lane & 15;        // M (A/C) or N (B/C) index within subtile
  const int hk   = lane >> 4;        // half-wave: K offset pair for A/B frags
  const int waveM = (wave >> 2) * 64;
  const int waveN = (wave & 3) * 32;

  const int blockN = blockIdx.x * TN;
  const int blockM = blockIdx.y * TM;

  // Global->LDS staging indices
  const int arow = tid >> 3;         // 0..31 (4 passes of 32 rows)
  const int acol = (tid & 7) * 4;    // 0..28
  const int brow = tid >> 5;         // 0..7  (4 passes of 8 k-rows)
  const int bcol = (tid & 31) * 4;   // 0..124

  v8f acc[4][2] = {};

  // prefetch first K-tile into registers
  f4 aPre[4], bPre[4];
#pragma unroll
  for (int i = 0; i < 4; ++i) {
    aPre[i] = *(const f4*)&A[(size_t)(blockM + arow + i * 32) * KTOT + acol];
    bPre[i] = *(const f4*)&B[(size_t)(brow + i * 8) * NTOT + blockN + bcol];
  }

  for (int kt = 0; kt < KTOT; kt += TK) {
    // commit prefetched tile to LDS
#pragma unroll
    for (int i = 0; i < 4; ++i) {
      *(f4*)&As[(arow + i * 32) * A_STRIDE + acol] = aPre[i];
      *(f4*)&Bs[(brow + i * 8) * B_STRIDE + bcol]  = bPre[i];
    }
    __syncthreads();

    // prefetch next tile while WMMAs run out of LDS
    const int ktn = kt + TK;
    if (ktn < KTOT) {
#pragma unroll
      for (int i = 0; i < 4; ++i) {
        aPre[i] = *(const f4*)&A[(size_t)(blockM + arow + i * 32) * KTOT + ktn + acol];
        bPre[i] = *(const f4*)&B[(size_t)(ktn + brow + i * 8) * NTOT + blockN + bcol];
      }
    }

#pragma unroll
    for (int kk = 0; kk < TK; kk += 4) {
      // A frag (16x4): lane holds A[m][kk + 2*hk .. +1] -> one b64 LDS read
      v2f afrag[4];
#pragma unroll
      for (int m = 0; m < 4; ++m)
        afrag[m] = *(const v2f*)&As[(waveM + m * 16 + lm) * A_STRIDE + kk + hk * 2];

      // B frag (4x16): lane holds B[kk + 2*hk][n], B[kk + 2*hk + 1][n]
      v2f bfrag[2];
#pragma unroll
      for (int n = 0; n < 2; ++n) {
        const int kb = (kk + hk * 2) * B_STRIDE + waveN + n * 16 + lm;
        bfrag[n].x = Bs[kb];
        bfrag[n].y = Bs[kb + B_STRIDE];
      }

#pragma unroll
      for (int m = 0; m < 4; ++m)
#pragma unroll
        for (int n = 0; n < 2; ++n)
          acc[m][n] = __builtin_amdgcn_wmma_f32_16x16x4_f32(
              false, afrag[m], false, bfrag[n], (short)0, acc[m][n],
              false, false);
    }
    __syncthreads();
  }

  // C layout: VGPR v -> row v (lanes 0-15) / row v+8 (lanes 16-31), col = lm
#pragma unroll
  for (int m = 0; m < 4; ++m) {
    const int row0 = blockM + waveM + m * 16 + hk * 8;
#pragma unroll
    for (int n = 0; n < 2; ++n) {
      const int col = blockN + waveN + n * 16 + lm;
#pragma unroll
      for (int v = 0; v < 8; ++v)
        C[(size_t)(row0 + v) * NTOT + col] = acc[m][n][v];
    }
  }
}

// ---------------------------------------------------------------------------
// SRU recurrence: one thread per (batch, hid) channel; coalesced streaming.
// ---------------------------------------------------------------------------
__device__ __forceinline__ float sigmoidf_(float z) {
  return 1.0f / (1.0f + __expf(-z));
}

__global__ __launch_bounds__(256) void sru_scan(
    const float* __restrict__ U, const float* __restrict__ xin,
    const float* __restrict__ bias, float* __restrict__ hout, int writeAll)
{
  const int idx = blockIdx.x * blockDim.x + threadIdx.x; // 0..BATCH*HID-1
  const int b = idx >> 10;         // / HID
  const int h = idx & (HID - 1);
  const float bf = bias[h];
  const float br = bias[HID + h];
  float c = 0.f;

  if (writeAll) {
    for (int t = 0; t < SEQ; ++t) {
      const size_t row = (size_t)t * BATCH + b;
      const float* Ur = U + row * NTOT;
      const float xt = Ur[h];
      const float f  = sigmoidf_(Ur[HID + h] + bf);
      c = f * c + (1.f - f) * xt;
      const float r  = sigmoidf_(Ur[2 * HID + h] + br);
      const float xv = xin[row * HID + h];
      hout[row * HID + h] = r * c + (1.f - r) * xv;
    }
  } else {
    for (int t = 0; t < SEQ; ++t) {
      const size_t row = (size_t)t * BATCH + b;
      const float* Ur = U + row * NTOT;
      const float xt = Ur[h];
      const float f  = sigmoidf_(Ur[HID + h] + bf);
      c = f * c + (1.f - f) * xt;
    }
    const size_t row = (size_t)(SEQ - 1) * BATCH + b;
    const float r  = sigmoidf_(U[row * NTOT + 2 * HID + h] + br);
    const float xv = xin[row * HID + h];
    hout[(size_t)b * HID + h] = r * c + (1.f - r) * xv;  // h[-1] only
  }
}

// ---------------------------------------------------------------------------
extern "C" void kernel_launch(void* const* d_in, const int* in_sizes, int n_in,
                              void* d_out, int out_size, void* d_ws, size_t ws_size,
                              hipStream_t stream) {
  const float* x  = (const float*)d_in[0];
  const float* W0 = (const float*)d_in[1];
  const float* b0 = (const float*)d_in[2];
  const float* W1 = (const float*)d_in[3];
  const float* b1 = (const float*)d_in[4];
  float* out = (float*)d_out;

  // workspace: U (32768 x 3072 f32 = 384 MB) then h1 (32768 x 1024 f32 = 128 MB)
  float* U  = (float*)d_ws;
  float* h1 = U + (size_t)MTOT * NTOT;

  const dim3 gemmGrid(NTOT / TN, MTOT / TM);   // 24 x 256 blocks
  const dim3 gemmBlk(256);
  const dim3 scanGrid((BATCH * HID) / 256);    // 64 blocks
  const dim3 scanBlk(256);

  // Layer 0
  wmma_gemm_f32<<<gemmGrid, gemmBlk, 0, stream>>>(x, W0, U);
  sru_scan<<<scanGrid, scanBlk, 0, stream>>>(U, x, b0, h1, 1);
  // Layer 1 (reuse U; only last timestep written to d_out)
  wmma_gemm_f32<<<gemmGrid, gemmBlk, 0, stream>>>(h1, W1, U);
  sru_scan<<<scanGrid, scanBlk, 0, stream>>>(U, h1, b1, out, 0);
}